// FaceDetector_22909355557572
// MI455X (gfx1250) — compile-verified
//
#include <hip/hip_runtime.h>
#include <stdint.h>
#include <math.h>

// ---------------- problem constants (from reference) ----------------
#define BATCH   8
#define NPRI    33600      // 160*160 + 80*80 + 40*40
#define SPAD    65536      // pow2 padding for bitonic sort
#define TOPK    5000
#define KEEPK   750
#define CONF_TH 0.3f
#define NMS_THR 0.3f
#define NT_NMS  1024
#define SORT_NT 1024
#define CHUNK   2048       // elements per LDS-resident sort chunk (16KB of u64 keys)

// ---------------- CDNA5 async global->LDS path ----------------
#if defined(__AMDGCN__) && __has_builtin(__builtin_amdgcn_global_load_async_to_lds_b128) && \
    __has_builtin(__builtin_amdgcn_global_load_async_to_lds_b32)
#define HAVE_ASYNC_LDS 1
#endif

// Builtin parameter types (confirmed by hipcc diagnostics):
//   b32 : (addrspace(1) int*, addrspace(3) int*, imm, imm)
//   b128: (addrspace(1) int __attribute__((vector_size(16)))*, addrspace(3) ..., imm, imm)
typedef int v4i_g __attribute__((vector_size(16)));

#define AS1_I32(p)  ((__attribute__((address_space(1))) int*)   (uintptr_t)(p))
#define AS1_V4I(p)  ((__attribute__((address_space(1))) v4i_g*) (uintptr_t)(p))
#define AS3_I32(p)  ((__attribute__((address_space(3))) int*)   (unsigned int)(uintptr_t)(p))
#define AS3_V4I(p)  ((__attribute__((address_space(3))) v4i_g*) (unsigned int)(uintptr_t)(p))

#ifdef HAVE_ASYNC_LDS
__device__ __forceinline__ void fd_wait_async0() {
#if __has_builtin(__builtin_amdgcn_s_wait_asynccnt)
    __builtin_amdgcn_s_wait_asynccnt(0);
#else
    asm volatile("s_wait_asynccnt 0" ::: "memory");
#endif
}
#endif

// ---------------- kernel 1: build sortable keys ----------------
// key = monotone(masked_score) << 32 | (0xFFFFFFFF - index)
// descending sort on key => score descending, ties by lowest index (JAX top_k).
__global__ void fd_score_key(const float* __restrict__ cls,
                             const float* __restrict__ obj,
                             unsigned long long* __restrict__ keys) {
    unsigned t = blockIdx.x * blockDim.x + threadIdx.x;   // BATCH*SPAD threads
    unsigned b = t >> 16;
    unsigned i = t & (SPAD - 1);
    unsigned long long key = 0ull;                         // padding sorts last
    if (i < NPRI) {
        size_t o = (size_t)b * NPRI + i;
        float s = sqrtf(cls[o] * obj[o]);
        float m = (s > CONF_TH) ? s : -1.0f;               // jnp.where(scores>TH, s, -1)
        unsigned u = __float_as_uint(m);
        u = ((int)u < 0) ? ~u : (u | 0x80000000u);         // order-preserving float->uint
        key = ((unsigned long long)u << 32) | (unsigned long long)(0xFFFFFFFFu - i);
    }
    keys[(size_t)b * SPAD + i] = key;
}

// ---------------- bitonic compare-swap helper (descending) ----------------
__device__ __forceinline__ void fd_cswap(unsigned long long* sk,
                                         unsigned i, unsigned ixj,
                                         unsigned gi, unsigned k) {
    unsigned long long a = sk[i];
    unsigned long long c = sk[ixj];
    bool swp = ((gi & k) == 0) ? (a < c) : (a > c);        // inverted => descending
    if (swp) { sk[i] = c; sk[ixj] = a; }
}

// ---------------- kernel 2a: global bitonic step (span j >= CHUNK) ----------------
__global__ void fd_bitonic(unsigned long long* __restrict__ keys, int k, int j) {
    unsigned t = blockIdx.x * blockDim.x + threadIdx.x;   // BATCH*SPAD threads
    unsigned b = t >> 16;
    unsigned i = t & (SPAD - 1);
    unsigned ixj = i ^ (unsigned)j;
    if (ixj > i) {
        unsigned long long* base = keys + (size_t)b * SPAD;
        unsigned long long a = base[i];
        unsigned long long c = base[ixj];
        bool swp = ((i & (unsigned)k) == 0) ? (a < c) : (a > c);
        if (swp) { base[i] = c; base[ixj] = a; }
    }
}

// ---------------- LDS chunk staging (async global->LDS) ----------------
__device__ __forceinline__ void fd_load_chunk(unsigned long long* sk,
                                              const unsigned long long* base,
                                              unsigned t) {
#ifdef HAVE_ASYNC_LDS
    // each lane pulls 16B (two u64 keys), coalesced and 16B-aligned
    __builtin_amdgcn_global_load_async_to_lds_b128(AS1_V4I(base + 2 * t),
                                                   AS3_V4I(&sk[2 * t]), 0, 0);
    fd_wait_async0();
#else
    sk[t]           = base[t];
    sk[t + SORT_NT] = base[t + SORT_NT];
#endif
    __syncthreads();
}

// ---------------- kernel 2b: LDS-resident full sort of 2048-chunks (k=2..CHUNK) ----------------
__global__ __launch_bounds__(SORT_NT, 1) void fd_bitonic_local_sort(
        unsigned long long* __restrict__ keys) {
    __shared__ unsigned long long sk[CHUNK];               // 16 KB
    unsigned chunk = blockIdx.x;                           // BATCH*SPAD/CHUNK blocks
    unsigned long long* base = keys + (size_t)chunk * CHUNK;
    unsigned t = threadIdx.x;
    unsigned gbase = (chunk * CHUNK) & (SPAD - 1);         // index within batch

    fd_load_chunk(sk, base, t);

    for (unsigned k = 2; k <= CHUNK; k <<= 1) {
        for (unsigned j = k >> 1; j > 0; j >>= 1) {
            unsigned i = ((t & ~(j - 1)) << 1) | (t & (j - 1));  // lower pair index
            fd_cswap(sk, i, i | j, gbase + i, k);
            __syncthreads();
        }
    }
    base[t]           = sk[t];
    base[t + SORT_NT] = sk[t + SORT_NT];
}

// ---------------- kernel 2c: LDS-resident merge tail (one k, j = CHUNK/2..1) ----------------
__global__ __launch_bounds__(SORT_NT, 1) void fd_bitonic_local_merge(
        unsigned long long* __restrict__ keys, int k) {
    __shared__ unsigned long long sk[CHUNK];
    unsigned chunk = blockIdx.x;
    unsigned long long* base = keys + (size_t)chunk * CHUNK;
    unsigned t = threadIdx.x;
    unsigned gbase = (chunk * CHUNK) & (SPAD - 1);

    fd_load_chunk(sk, base, t);

    for (unsigned j = CHUNK >> 1; j > 0; j >>= 1) {
        unsigned i = ((t & ~(j - 1)) << 1) | (t & (j - 1));
        fd_cswap(sk, i, i | j, gbase + i, (unsigned)k);
        __syncthreads();
    }
    base[t]           = sk[t];
    base[t + SORT_NT] = sk[t + SORT_NT];
}

// ---------------- kernel 3: gather top-K + decode ----------------
__global__ void fd_gather_decode(const unsigned long long* __restrict__ keys,
                                 const float* __restrict__ cls,
                                 const float* __restrict__ obj,
                                 const float* __restrict__ bbox,
                                 const float* __restrict__ kps,
                                 const float* __restrict__ priors,
                                 float*  __restrict__ topdets,   // [B*TOPK][15]
                                 float4* __restrict__ boxarr,    // [B*TOPK] x1y1x2y2
                                 float*  __restrict__ scorearr)  // [B*TOPK]
{
    int t = blockIdx.x * blockDim.x + threadIdx.x;
    if (t >= BATCH * TOPK) return;
    int b = t / TOPK;
    int r = t - b * TOPK;
    unsigned long long key = keys[(size_t)b * SPAD + r];
    int idx = (int)(0xFFFFFFFFu - (unsigned)(key & 0xFFFFFFFFull));
    if ((unsigned)idx >= (unsigned)NPRI) idx = 0;          // safety (padding can't reach top-K)

    float px = priors[idx * 4 + 0];
    float py = priors[idx * 4 + 1];
    float pw = priors[idx * 4 + 2];
    float ph = priors[idx * 4 + 3];
    size_t base = (size_t)b * NPRI + idx;

    const float* bp = bbox + base * 4;
    float cx = px + bp[0] * pw;
    float cy = py + bp[1] * ph;
    float w  = pw * expf(bp[2]);
    float h  = ph * expf(bp[3]);
    float x1 = cx - w * 0.5f;
    float y1 = cy - h * 0.5f;
    float x2 = x1 + w;
    float y2 = y1 + h;
    float s  = sqrtf(cls[base] * obj[base]);

    float* row = topdets + (size_t)t * 15;
    row[0] = x1; row[1] = y1; row[2] = x2; row[3] = y2;
    const float* kp = kps + base * 10;
#pragma unroll
    for (int q = 0; q < 5; ++q) {
        row[4 + 2 * q]     = px + kp[2 * q]     * pw;
        row[4 + 2 * q + 1] = py + kp[2 * q + 1] * ph;
    }
    row[14] = s;

    boxarr[t]   = make_float4(x1, y1, x2, y2);
    scorearr[t] = s;
}

// ---------------- kernel 4: LDS-resident greedy NMS + final gather ----------------
__global__ __launch_bounds__(NT_NMS, 1) void fd_nms_finalize(
        const float4* __restrict__ boxarr,
        const float*  __restrict__ scorearr,
        const float*  __restrict__ topdets,
        float* __restrict__ out,        // [B][KEEPK][15]
        float* __restrict__ outvalid)   // [B][KEEPK]
{
    __shared__ float4        sbox[TOPK];     // 80 KB
    __shared__ float         sscore[TOPK];   // 20 KB
    __shared__ unsigned char skeep[TOPK];    //  5 KB
    __shared__ int           ord[KEEPK];
    __shared__ int           snkeep;

    int b = blockIdx.x;
    int t = threadIdx.x;
    const float4* gbox   = boxarr  + (size_t)b * TOPK;
    const float*  gscore = scorearr + (size_t)b * TOPK;

    for (int i = t; i < TOPK; i += NT_NMS) {
#ifdef HAVE_ASYNC_LDS
        __builtin_amdgcn_global_load_async_to_lds_b128(AS1_V4I(gbox + i),   AS3_V4I(&sbox[i]),   0, 0);
        __builtin_amdgcn_global_load_async_to_lds_b32 (AS1_I32(gscore + i), AS3_I32(&sscore[i]), 0, 0);
#else
        sbox[i]   = gbox[i];
        sscore[i] = gscore[i];
#endif
    }
#ifdef HAVE_ASYNC_LDS
    fd_wait_async0();
#endif
    __syncthreads();

    for (int i = t; i < TOPK; i += NT_NMS)
        skeep[i] = (sscore[i] > CONF_TH) ? 1 : 0;          // valid = top_scores > TH
    __syncthreads();

    // Greedy suppression identical to the reference fori_loop.
    for (int i = 0; i < TOPK; ++i) {
        if (skeep[i]) {                                    // uniform branch (LDS scalar)
            float4 bi = sbox[i];
            float ai = fmaxf(bi.z - bi.x, 0.0f) * fmaxf(bi.w - bi.y, 0.0f);
            for (int j = i + 1 + t; j < TOPK; j += NT_NMS) {
                if (!skeep[j]) continue;
                float4 bj = sbox[j];
                float aj = fmaxf(bj.z - bj.x, 0.0f) * fmaxf(bj.w - bj.y, 0.0f);
                float lx = fmaxf(bi.x, bj.x), ly = fmaxf(bi.y, bj.y);
                float rx = fminf(bi.z, bj.z), ry = fminf(bi.w, bj.w);
                float iw = fmaxf(rx - lx, 0.0f), ih = fmaxf(ry - ly, 0.0f);
                float inter = iw * ih;
                float iou = inter / (ai + aj - inter + 1e-12f);
                if (iou > NMS_THR) skeep[j] = 0;
            }
        }
        __syncthreads();
    }

    // Stable partition: kept indices (ascending), then non-kept (ascending) ==
    // top_k(where(keep, sorted_scores, -1), 750) ordering.
    if (t == 0) {
        int p = 0;
        for (int i = 0; i < TOPK && p < KEEPK; ++i) if (skeep[i])  ord[p++] = i;
        snkeep = p;
        for (int i = 0; i < TOPK && p < KEEPK; ++i) if (!skeep[i]) ord[p++] = i;
    }
    __syncthreads();

    int nk = snkeep;
    const float* drows = topdets + (size_t)b * TOPK * 15;
    for (int r = t; r < KEEPK; r += NT_NMS) {
        int src = ord[r];
        const float* drow = drows + (size_t)src * 15;
        float* orow = out + ((size_t)b * KEEPK + r) * 15;
#pragma unroll
        for (int c = 0; c < 15; ++c) orow[c] = drow[c];
        outvalid[b * KEEPK + r] = (r < nk) ? 1.0f : 0.0f;
    }
}

// ---------------- host launcher ----------------
extern "C" void kernel_launch(void* const* d_in, const int* in_sizes, int n_in,
                              void* d_out, int out_size, void* d_ws, size_t ws_size,
                              hipStream_t stream) {
    const float* cls    = (const float*)d_in[0];   // (B,N,1)
    const float* obj    = (const float*)d_in[1];   // (B,N,1)
    const float* bbox   = (const float*)d_in[2];   // (B,N,4)
    const float* kps    = (const float*)d_in[3];   // (B,N,10)
    const float* priors = (const float*)d_in[4];   // (N,4)

    char* ws = (char*)d_ws;
    size_t off = 0;
    unsigned long long* keys = (unsigned long long*)(ws + off); off += (size_t)BATCH * SPAD * 8;  // 4,194,304
    float*  topdets  = (float*) (ws + off); off += (size_t)BATCH * TOPK * 15 * 4;                 // 2,400,000
    float4* boxarr   = (float4*)(ws + off); off += (size_t)BATCH * TOPK * 16;                     //   640,000 (16B aligned)
    float*  scorearr = (float*) (ws + off); off += (size_t)BATCH * TOPK * 4;                      //   160,000
    (void)ws_size; (void)in_sizes; (void)n_in; (void)out_size;

    float* out      = (float*)d_out;                    // (B,750,15)
    float* outvalid = out + (size_t)BATCH * KEEPK * 15; // (B,750)

    const int nthr    = BATCH * SPAD;                   // 524288
    const int nchunks = nthr / CHUNK;                   // 256

    fd_score_key<<<nthr / 256, 256, 0, stream>>>(cls, obj, keys);

    // LDS-fused bitonic sort: 21 launches instead of 136.
    fd_bitonic_local_sort<<<nchunks, SORT_NT, 0, stream>>>(keys);        // all k <= CHUNK
    for (unsigned k = 2 * CHUNK; k <= SPAD; k <<= 1) {
        for (unsigned j = k >> 1; j >= CHUNK; j >>= 1)                   // spans crossing chunks
            fd_bitonic<<<nthr / 256, 256, 0, stream>>>(keys, (int)k, (int)j);
        fd_bitonic_local_merge<<<nchunks, SORT_NT, 0, stream>>>(keys, (int)k);  // j <= CHUNK/2
    }

    fd_gather_decode<<<(BATCH * TOPK + 255) / 256, 256, 0, stream>>>(
        keys, cls, obj, bbox, kps, priors, topdets, boxarr, scorearr);

    fd_nms_finalize<<<BATCH, NT_NMS, 0, stream>>>(boxarr, scorearr, topdets, out, outvalid);
}